// RoiHead_56092272886452
// MI455X (gfx1250) — compile-verified
//
#include <hip/hip_runtime.h>
#include <float.h>

// ROI max-pool: features (2,256,50,50) f32, proposals (2,128,4) i32.
// One workgroup per ROI. Double-buffered async global->LDS staging (ASYNCcnt)
// overlapped with per-wave shfl-xor max reductions. Division-free addressing:
// wave -> channel, uniform loop -> row, lane -> column.

#define BLOCK_THREADS 256
#define CH_PER_CHUNK  8                  // == waves per block; 1 channel/wave
#define NCHUNK        (CHANS / CH_PER_CHUNK)   // 32
#define MAX_REGION    576                // coords < 25 -> w,h <= 24 -> 24*24
#define CHANS         256
#define HDIM          50
#define WDIM          50
#define PLANE         (HDIM * WDIM)      // 2500

__global__ __launch_bounds__(BLOCK_THREADS)
void roi_maxpool_async_kernel(const float* __restrict__ features,
                              const int*   __restrict__ proposals,
                              float*       __restrict__ out)
{
    __shared__ float smem[2][CH_PER_CHUNK * MAX_REGION];   // 36 KB total

    const int roi  = blockIdx.x;          // 0..255 ; batch-major (b*128 + p)
    const int b    = roi >> 7;            // 128 proposals per batch
    const int lane = threadIdx.x & 31;
    const int wave = threadIdx.x >> 5;    // 0..7 -> channel within chunk

    // proposal -> feature-grid coords (// FEATURE_STRIDE == >> 4, values > 0)
    const int4 pr = ((const int4*)proposals)[roi];
    const int x = pr.x >> 4;
    const int y = pr.y >> 4;
    const int w = pr.z >> 4;              // 1..24  (< wave32 width)
    const int h = pr.w >> 4;              // 1..24
    const int hw = w * h;                 // 1..576, region fully in-bounds

    const float* fb = features + (size_t)b * CHANS * PLANE;     // uniform -> SGPR
    const unsigned lbase = (unsigned)(size_t)(const void*)smem; // LDS byte addr

    // Issue one chunk's region (8 channels) as async global->LDS copies.
    // Per wave: h wave-level issues, EXEC = (lane < w); consecutive lanes read
    // consecutive floats of a row -> coalesced. Packed LDS layout: cl*hw + r*w + col.
    auto issue_chunk = [&](int chunk, int buf) {
        if (lane < w) {
            const int gchan = (chunk * CH_PER_CHUNK + wave) * PLANE;
            const unsigned lchan = lbase +
                (((unsigned)(buf * (CH_PER_CHUNK * MAX_REGION) + wave * hw + lane)) << 2);
            for (int r = 0; r < h; ++r) {            // uniform trip count
                const int goff = (gchan + (y + r) * WDIM + x + lane) << 2;
                const unsigned laddr = lchan + ((unsigned)(r * w) << 2);
                // GVS mode: LDS-dest VGPR, 32-bit byte-offset VGPR, 64-bit SGPR base
                asm volatile("global_load_async_to_lds_b32 %0, %1, %2"
                             :: "v"(laddr), "v"(goff), "s"(fb)
                             : "memory");
            }
        }
    };

    issue_chunk(0, 0);                                // prime the pipeline

    for (int k = 0; k < NCHUNK; ++k) {
        const int buf = k & 1;

        asm volatile("s_wait_asynccnt 0" ::: "memory");  // chunk k landed (this wave)
        __syncthreads();                                 // ... and all waves' copies

        if (k + 1 < NCHUNK)
            issue_chunk(k + 1, buf ^ 1);  // async copies overlap compute below

        // max-reduce this wave's channel out of LDS
        float m = -FLT_MAX;               // == jnp.finfo(f32).min fill value
        for (int e = lane; e < hw; e += 32)
            m = fmaxf(m, smem[buf][wave * hw + e]);
        #pragma unroll
        for (int off = 16; off > 0; off >>= 1)
            m = fmaxf(m, __shfl_xor(m, off, 32));
        if (lane == 0)
            out[roi * CHANS + k * CH_PER_CHUNK + wave] = m;

        __syncthreads();                  // all reads of buf done before chunk k+2 reuses it
    }
}

extern "C" void kernel_launch(void* const* d_in, const int* in_sizes, int n_in,
                              void* d_out, int out_size, void* d_ws, size_t ws_size,
                              hipStream_t stream) {
    (void)n_in; (void)out_size; (void)d_ws; (void)ws_size;
    const float* features  = (const float*)d_in[0];   // (2,256,50,50) f32
    const int*   proposals = (const int*)d_in[1];     // (2,128,4) i32
    float*       out       = (float*)d_out;           // (256,256,1,1) f32

    const int n_roi = in_sizes[1] / 4;                // 256
    roi_maxpool_async_kernel<<<n_roi, BLOCK_THREADS, 0, stream>>>(features, proposals, out);
}